// IntegralLoss_Quadratic_24996709663348
// MI455X (gfx1250) — compile-verified
//
#include <hip/hip_runtime.h>

typedef __attribute__((ext_vector_type(2))) float v2f;
typedef __attribute__((ext_vector_type(8))) float v8f;

#define TPB 256

// out[i] = || M * [P_i ; chart6(inv(Hcs * exp(q_i) * Hid))] + c ||_2
//   M = [ Kd@I_inv | Ad^T@Kp ] (6x12),  c = Ad^T @ (W_grv_real - W_grv)
__global__ __launch_bounds__(TPB) void se3_loss_kernel(
    const float* __restrict__ x,
    const float* __restrict__ I_inv,
    const float* __restrict__ Kd,
    const float* __restrict__ Kp,
    const float* __restrict__ Hcs,
    const float* __restrict__ Hid,
    const float* __restrict__ Ad,
    const float* __restrict__ Wg,
    const float* __restrict__ Wgr,
    float* __restrict__ out)
{
    __shared__ __align__(16) float xs[TPB * 13];   // staged x tile
    __shared__ __align__(16) float ys[TPB * 12];   // per-element y = [P ; g]
    __shared__ __align__(16) float Ms[256 + 8];    // Mpad(16x16) + c(6)

    const int tid  = threadIdx.x;
    const int lane = tid & 31;
    const int wave = tid >> 5;
    const int m16  = lane & 15;
    const int hh   = lane >> 4;           // K-half selector for WMMA operands
    const size_t blockBase = (size_t)blockIdx.x * TPB;

    // ---- stage x tile (256 elems * 13 floats) with coalesced b32 loads ----
    const float* gx = x + blockBase * 13;
#pragma unroll
    for (int j = 0; j < 13; ++j)
        xs[tid + TPB * j] = gx[tid + TPB * j];

    // ---- fold constant matrices into Mpad (rows>=6 / cols>=12 are zero) ----
    {
        const int m = tid >> 4, k = tid & 15;
        float val = 0.f;
        if (m < 6) {
            if (k < 6) {
                float s = 0.f;
#pragma unroll
                for (int j = 0; j < 6; ++j) s += Kd[m*6 + j] * I_inv[j*6 + k];
                val = s;
            } else if (k < 12) {
                const int kk = k - 6;
                float s = 0.f;
#pragma unroll
                for (int j = 0; j < 6; ++j) s += Ad[j*6 + m] * Kp[j*6 + kk];
                val = s;
            }
        }
        Ms[tid] = val;
        if (tid < 6) {
            float s = 0.f;
#pragma unroll
            for (int j = 0; j < 6; ++j) s += Ad[j*6 + tid] * (Wgr[j] - Wg[j]);
            Ms[256 + tid] = s;
        }
    }

    // ---- constant SE(3) frames (uniform; L2-resident) ----
    float Ra[9], Rc[9], pa[3], pc[3];
#pragma unroll
    for (int i = 0; i < 3; ++i) {
#pragma unroll
        for (int j = 0; j < 3; ++j) { Ra[i*3+j] = Hcs[i*4+j]; Rc[i*3+j] = Hid[i*4+j]; }
        pa[i] = Hcs[i*4+3];
        pc[i] = Hid[i*4+3];
    }

    __syncthreads();

    // ---- WMMA A operands: 16x4 f32 slices of Mpad, lane-swizzled ----
    // A layout (f32 16x4): lane = M row (mod 16); vgpr0 -> K = 0 + 2*hh, vgpr1 -> K = 1 + 2*hh
    v2f a0 = *(const v2f*)(Ms + m16*16 + 0 + 2*hh);
    v2f a1 = *(const v2f*)(Ms + m16*16 + 4 + 2*hh);
    v2f a2 = *(const v2f*)(Ms + m16*16 + 8 + 2*hh);
    v8f cinit;
    cinit[0] = Ms[256+0]; cinit[1] = Ms[256+1]; cinit[2] = Ms[256+2];
    cinit[3] = Ms[256+3]; cinit[4] = Ms[256+4]; cinit[5] = Ms[256+5];
    cinit[6] = 0.f;       cinit[7] = 0.f;

    // ---- per-element SE(3) exponential + frame chain (VALU) ----
    const float* xe = xs + tid * 13;
    const float wx = xe[0], wy = xe[1], wz = xe[2];
    const float vx = xe[3], vy = xe[4], vz = xe[5];

    const float th2  = wx*wx + wy*wy + wz*wz;
    const bool  sml  = th2 < 1e-8f;
    const float th2s = sml ? 1.0f : th2;
    const float th   = __builtin_amdgcn_sqrtf(th2s);   // hw v_sqrt_f32 (~1 ulp)
    float sn, cs;
    __sincosf(th, &sn, &cs);
    const float Af = sml ? 1.f  - th2 * (1.f/6.f)   : __fdividef(sn, th);
    const float Bf = sml ? 0.5f - th2 * (1.f/24.f)  : __fdividef(1.f - cs, th2s);
    const float Cf = sml ? (1.f/6.f) - th2*(1.f/120.f)
                         : __fdividef(th - sn, th2s * th);

    // W^2 = w w^T - th2*I  (skew-square identity)
    const float xx = wx*wx, yy = wy*wy, zz = wz*wz;
    const float xy = wx*wy, xz = wx*wz, yz = wy*wz;

    // Rx = I + Af*W + Bf*W^2
    const float r00 = 1.f + Bf*(xx - th2);
    const float r01 = Bf*xy - Af*wz;
    const float r02 = Bf*xz + Af*wy;
    const float r10 = Bf*xy + Af*wz;
    const float r11 = 1.f + Bf*(yy - th2);
    const float r12 = Bf*yz - Af*wx;
    const float r20 = Bf*xz - Af*wy;
    const float r21 = Bf*yz + Af*wx;
    const float r22 = 1.f + Bf*(zz - th2);

    // V = I + Bf*W + Cf*W^2 ;  px = V*v
    const float q00 = 1.f + Cf*(xx - th2);
    const float q01 = Cf*xy - Bf*wz;
    const float q02 = Cf*xz + Bf*wy;
    const float q10 = Cf*xy + Bf*wz;
    const float q11 = 1.f + Cf*(yy - th2);
    const float q12 = Cf*yz - Bf*wx;
    const float q20 = Cf*xz - Bf*wy;
    const float q21 = Cf*yz + Bf*wx;
    const float q22 = 1.f + Cf*(zz - th2);
    const float px0 = q00*vx + q01*vy + q02*vz;
    const float px1 = q10*vx + q11*vy + q12*vz;
    const float px2 = q20*vx + q21*vy + q22*vz;

    // T1 = Rx * Rc
    const float t00 = r00*Rc[0] + r01*Rc[3] + r02*Rc[6];
    const float t01 = r00*Rc[1] + r01*Rc[4] + r02*Rc[7];
    const float t02 = r00*Rc[2] + r01*Rc[5] + r02*Rc[8];
    const float t10 = r10*Rc[0] + r11*Rc[3] + r12*Rc[6];
    const float t11 = r10*Rc[1] + r11*Rc[4] + r12*Rc[7];
    const float t12 = r10*Rc[2] + r11*Rc[5] + r12*Rc[8];
    const float t20 = r20*Rc[0] + r21*Rc[3] + r22*Rc[6];
    const float t21 = r20*Rc[1] + r21*Rc[4] + r22*Rc[7];
    const float t22 = r20*Rc[2] + r21*Rc[5] + r22*Rc[8];

    // R = Ra * T1
    const float R00 = Ra[0]*t00 + Ra[1]*t10 + Ra[2]*t20;
    const float R01 = Ra[0]*t01 + Ra[1]*t11 + Ra[2]*t21;
    const float R02 = Ra[0]*t02 + Ra[1]*t12 + Ra[2]*t22;
    const float R10 = Ra[3]*t00 + Ra[4]*t10 + Ra[5]*t20;
    const float R11 = Ra[3]*t01 + Ra[4]*t11 + Ra[5]*t21;
    const float R12 = Ra[3]*t02 + Ra[4]*t12 + Ra[5]*t22;
    const float R20 = Ra[6]*t00 + Ra[7]*t10 + Ra[8]*t20;
    const float R21 = Ra[6]*t01 + Ra[7]*t11 + Ra[8]*t21;
    const float R22 = Ra[6]*t02 + Ra[7]*t12 + Ra[8]*t22;

    // p = Ra*(Rx*pc + px) + pa
    const float pt0 = r00*pc[0] + r01*pc[1] + r02*pc[2] + px0;
    const float pt1 = r10*pc[0] + r11*pc[1] + r12*pc[2] + px1;
    const float pt2 = r20*pc[0] + r21*pc[1] + r22*pc[2] + px2;
    const float p0  = Ra[0]*pt0 + Ra[1]*pt1 + Ra[2]*pt2 + pa[0];
    const float p1  = Ra[3]*pt0 + Ra[4]*pt1 + Ra[5]*pt2 + pa[1];
    const float p2  = Ra[6]*pt0 + Ra[7]*pt1 + Ra[8]*pt2 + pa[2];

    // g = chart6(inv(H)) computed directly from H
    const float gw0 = 0.5f*(R12 - R21);
    const float gw1 = 0.5f*(R20 - R02);
    const float gw2 = 0.5f*(R01 - R10);
    const float gv0 = -(R00*p0 + R10*p1 + R20*p2);
    const float gv1 = -(R01*p0 + R11*p1 + R21*p2);
    const float gv2 = -(R02*p0 + R12*p1 + R22*p2);

    float* ye = ys + tid * 12;
    ye[0] = xe[6];  ye[1] = xe[7];  ye[2]  = xe[8];
    ye[3] = xe[9];  ye[4] = xe[10]; ye[5]  = xe[11];
    ye[6] = gw0;    ye[7] = gw1;    ye[8]  = gw2;
    ye[9] = gv0;    ye[10] = gv1;   ye[11] = gv2;

    __syncthreads();

    // ---- matrix pipe: u(16 elems) = Mpad * y + c, two 16-elem tiles per wave ----
    // B layout (f32 4x16): lane = column N (mod 16); vgpr0 -> K = 0 + 2*hh, vgpr1 -> K = 1 + 2*hh
    float res0 = 0.f, res1 = 0.f;
#pragma unroll
    for (int t = 0; t < 2; ++t) {
        const float* yb = ys + (size_t)(wave*32 + t*16 + m16) * 12 + 2*hh;
        v2f b0 = *(const v2f*)(yb + 0);
        v2f b1 = *(const v2f*)(yb + 4);
        v2f b2 = *(const v2f*)(yb + 8);
        v8f acc = cinit;
        acc = __builtin_amdgcn_wmma_f32_16x16x4_f32(false, a0, false, b0, (short)0, acc, false, false);
        acc = __builtin_amdgcn_wmma_f32_16x16x4_f32(false, a1, false, b1, (short)0, acc, false, false);
        acc = __builtin_amdgcn_wmma_f32_16x16x4_f32(false, a2, false, b2, (short)0, acc, false, false);
        // D layout: lane n (0..15) holds u[m] in acc[m], m = 0..5 valid
        const float ss = acc[0]*acc[0] + acc[1]*acc[1] + acc[2]*acc[2]
                       + acc[3]*acc[3] + acc[4]*acc[4] + acc[5]*acc[5];
        const float r = __builtin_amdgcn_sqrtf(ss);    // hw v_sqrt_f32
        if (t == 0) res0 = r; else res1 = r;
    }

    if (lane < 16) {
        const size_t ob = blockBase + (size_t)wave * 32;
        out[ob + m16]      = res0;
        out[ob + 16 + m16] = res1;
    }
}

extern "C" void kernel_launch(void* const* d_in, const int* in_sizes, int n_in,
                              void* d_out, int out_size, void* d_ws, size_t ws_size,
                              hipStream_t stream) {
    (void)n_in; (void)out_size; (void)d_ws; (void)ws_size;
    // setup_inputs order: t, x, I_inv, Kd, Kp, H_CS_B, H_I_D, Ad_CS_B, W_grv, W_grv_real
    const float* x    = (const float*)d_in[1];
    const float* Iinv = (const float*)d_in[2];
    const float* Kd   = (const float*)d_in[3];
    const float* Kp   = (const float*)d_in[4];
    const float* Hcs  = (const float*)d_in[5];
    const float* Hid  = (const float*)d_in[6];
    const float* Ad   = (const float*)d_in[7];
    const float* Wg   = (const float*)d_in[8];
    const float* Wgr  = (const float*)d_in[9];
    float* out = (float*)d_out;

    const int B = in_sizes[1] / 13;        // 2097152, divisible by TPB
    const int grid = B / TPB;
    se3_loss_kernel<<<grid, TPB, 0, stream>>>(x, Iinv, Kd, Kp, Hcs, Hid, Ad, Wg, Wgr, out);
}